// LIFNeuron_56221121904851
// MI455X (gfx1250) — compile-verified
//
#include <hip/hip_runtime.h>
#include <cstdint>
#include <math.h>

// LIF neuron multi-step scan for MI455X (gfx1250).
// Bandwidth-bound streaming scan: 64 MB in, 128 MB out -> ~8.2 us floor @ 23.3 TB/s.
// CDNA5 async global->LDS copies (ASYNCcnt) form an 8-deep read pipeline
// (2048 waves x 8 x 512 B = 8 MB reads in flight, covering ~5.4 MB needed to
// sustain the 7.8 TB/s read share at ~700 ns latency). Non-temporal hints on
// both the read-once input and write-once outputs keep the 192 MB L2 clean.
// No WMMA: elementwise temporal scan has no matrix contraction.

typedef __attribute__((ext_vector_type(4))) float v4f;
#define AS3 __attribute__((address_space(3)))

constexpr int   T_STEPS = 64;     // reference: T = 64 (fixed)
constexpr int   BLOCK   = 256;    // 8 waves of 32
constexpr int   DEPTH   = 8;      // async pipeline depth (power of two, <= T)
constexpr int   VEC     = 4;      // floats per lane per step (b128)
constexpr float V_TH    = 1.0f;

static_assert((DEPTH & (DEPTH - 1)) == 0, "DEPTH must be power of two");
static_assert(DEPTH <= T_STEPS, "pipeline deeper than sequence");

__device__ __forceinline__ void async_load_b128(unsigned lds_byte_off, const void* gptr) {
    // CDNA5: GLOBAL_LOAD_ASYNC_TO_LDS_B128 (tracked by ASYNCcnt).
    // vdst = VGPR holding LDS byte address, vaddr = 64-bit global address, saddr = off.
    // NT hint: input is read exactly once; don't retain it in the cache hierarchy.
    asm volatile("global_load_async_to_lds_b128 %0, %1, off th:TH_LOAD_NT"
                 :: "v"(lds_byte_off), "v"((unsigned long long)(uintptr_t)gptr)
                 : "memory");
}

// Wait until ASYNCcnt <= n. The immediate must be literal; switch folds when n
// is compile-time constant (unrolled epilogue) and is tiny otherwise.
__device__ __forceinline__ void wait_async_le(int n) {
    switch (n) {
        case 0:  asm volatile("s_wait_asynccnt 0x0" ::: "memory"); break;
        case 1:  asm volatile("s_wait_asynccnt 0x1" ::: "memory"); break;
        case 2:  asm volatile("s_wait_asynccnt 0x2" ::: "memory"); break;
        case 3:  asm volatile("s_wait_asynccnt 0x3" ::: "memory"); break;
        case 4:  asm volatile("s_wait_asynccnt 0x4" ::: "memory"); break;
        case 5:  asm volatile("s_wait_asynccnt 0x5" ::: "memory"); break;
        case 6:  asm volatile("s_wait_asynccnt 0x6" ::: "memory"); break;
        default: asm volatile("s_wait_asynccnt 0x7" ::: "memory"); break;
    }
}

__global__ __launch_bounds__(BLOCK) void lif_scan_async(
    const float* __restrict__ x,        // [T, Nplane]
    float* __restrict__ spike_out,      // [T, Nplane]
    float* __restrict__ mem_out,        // [T, Nplane]
    int Nplane)
{
    __shared__ v4f lds[DEPTH][BLOCK];   // 32 KB of the 320 KB/WGP pool

    const int tid = threadIdx.x;
    const long long base = (long long)blockIdx.x * (BLOCK * VEC) + tid * VEC;

    // LDS byte offset of this lane's slice in buffer 0 (addrspacecast -> 32-bit).
    const unsigned lds0 = (unsigned)(uintptr_t)(AS3 v4f*)&lds[0][tid];
    constexpr unsigned BUF_STRIDE = BLOCK * sizeof(v4f);  // 4096 B per buffer

    const float beta = expf(-1.0f / (2.0f + 1e-6f));      // exp(-DT/(TAU_MEM+EPS))

    // Prologue: fill the pipeline with loads for t = 0..DEPTH-1.
    #pragma unroll
    for (int t = 0; t < DEPTH; ++t)
        async_load_b128(lds0 + (unsigned)t * BUF_STRIDE,
                        x + (long long)t * Nplane + base);

    v4f mem = {};  // v_rest = 0

    auto consume = [&](int t) {
        // Each lane reads back only its own 16 B slice -> no barrier needed;
        // per-wave ASYNCcnt ordering is sufficient.
        const v4f xv = lds[t & (DEPTH - 1)][tid];
        v4f sp;
        #pragma unroll
        for (int c = 0; c < 4; ++c) {
            float m = fmaf(mem[c], beta, xv[c]);
            const bool fire = (m >= V_TH);          // H(mem - v_th)
            sp[c]  = fire ? 1.0f : 0.0f;
            mem[c] = fire ? 0.0f : m;               // hard reset, v_reset = 0
        }
        const long long o = (long long)t * Nplane + base;
        __builtin_nontemporal_store(sp,  (v4f*)(spike_out + o));
        __builtin_nontemporal_store(mem, (v4f*)(mem_out  + o));
    };

    // Main loop: ASYNCcnt <= DEPTH-1 with in-order completion => load t retired.
    // Consume buffer t, then refill it with load t+DEPTH.
    for (int t = 0; t < T_STEPS - DEPTH; ++t) {
        wait_async_le(DEPTH - 1);
        consume(t);
        async_load_b128(lds0 + (unsigned)(t & (DEPTH - 1)) * BUF_STRIDE,
                        x + (long long)(t + DEPTH) * Nplane + base);
    }

    // Epilogue: drain the last DEPTH buffers with progressively tighter waits so
    // earlier buffers' compute/stores overlap the still-in-flight later loads.
    #pragma unroll
    for (int k = 0; k < DEPTH; ++k) {
        wait_async_le(DEPTH - 1 - k);           // constant-folds per iteration
        consume(T_STEPS - DEPTH + k);
    }
}

// Fallback (plane size not a multiple of BLOCK*VEC): plain scalar streaming scan.
__global__ void lif_scan_scalar(const float* __restrict__ x,
                                float* __restrict__ spike_out,
                                float* __restrict__ mem_out,
                                int Nplane, int T)
{
    const long long i = (long long)blockIdx.x * blockDim.x + threadIdx.x;
    if (i >= Nplane) return;
    const float beta = expf(-1.0f / (2.0f + 1e-6f));
    float m = 0.0f;
    for (int t = 0; t < T; ++t) {
        const long long o = (long long)t * Nplane + i;
        const float xv = __builtin_nontemporal_load(x + o);
        m = fmaf(m, beta, xv);
        const bool fire = (m >= V_TH);
        __builtin_nontemporal_store(fire ? 1.0f : 0.0f, spike_out + o);
        m = fire ? 0.0f : m;
        __builtin_nontemporal_store(m, mem_out + o);
    }
}

extern "C" void kernel_launch(void* const* d_in, const int* in_sizes, int n_in,
                              void* d_out, int out_size, void* d_ws, size_t ws_size,
                              hipStream_t stream)
{
    const float* x   = (const float*)d_in[0];
    float*       out = (float*)d_out;

    const int total  = in_sizes[0];          // T * B * F
    const int Nplane = total / T_STEPS;      // B * F elements per time step

    float* spike = out;                      // output 0: spike_seq [T, B, F]
    float* memsq = out + total;              // output 1: mem_seq   [T, B, F]

    if (Nplane % (BLOCK * VEC) == 0) {
        const int blocks = Nplane / (BLOCK * VEC);
        lif_scan_async<<<blocks, BLOCK, 0, stream>>>(x, spike, memsq, Nplane);
    } else {
        const int blocks = (Nplane + BLOCK - 1) / BLOCK;
        lif_scan_scalar<<<blocks, BLOCK, 0, stream>>>(x, spike, memsq, Nplane, T_STEPS);
    }
}